// Downsample_67456756351403
// MI455X (gfx1250) — compile-verified
//
#include <hip/hip_runtime.h>

// ---------------------------------------------------------------------------
// Furthest Point Sampling for MI455X (gfx1250, wave32).
// One 1024-thread workgroup per batch (32 waves, 8 per SIMD32).
// Per-point running min-distances live in VGPRs (64 per lane).
// xyz streamed per iteration through a wave-private async global->LDS
// pipeline: 4 LDS buffers, 3 chunks (9 async b128 ops) in flight,
// wave-local s_wait_asynccnt only -- no barriers in the stream loop.
// Blocked point ownership (lane*4+jj per chunk) => the consume is
// 3 x ds_load_b128 per lane per chunk (48B contiguous, 16B aligned).
// Argmax uses packed (value|~idx) u64 keys: one 64-bit max == argmax with
// lowest-index tie-break (matches jnp.argmax).
// ---------------------------------------------------------------------------

constexpr int kN = 65536;           // points per batch
constexpr int kM = 2048;            // centers to select
constexpr int kT = 1024;            // threads per block
constexpr int kW = kT / 32;         // 32 waves
constexpr int kPtsPerWave = kN / kW;        // 2048 (== 1 << 11)
constexpr int kJ = kPtsPerWave / 32;        // 64 points per lane
constexpr int kChunkJ = 4;                  // points per lane per chunk
constexpr int kChunkPts = kChunkJ * 32;     // 128 points per chunk
constexpr int kChunks = kJ / kChunkJ;       // 16 chunks
constexpr int kChunkBytes = kChunkPts * 12;         // 1536 B per chunk
constexpr int kBuf = 4;                             // pipeline depth
constexpr int kStageBytes = kW * kBuf * kChunkBytes;  // 192 KB

#if defined(__has_builtin)
#if __has_builtin(__builtin_amdgcn_global_load_async_to_lds_b128)
#define HAVE_ASYNC_B128 1
#endif
#if __has_builtin(__builtin_amdgcn_s_wait_asynccnt)
#define HAVE_WAIT_ASYNC 1
#endif
#endif
#ifndef HAVE_ASYNC_B128
#define HAVE_ASYNC_B128 0
#endif
#ifndef HAVE_WAIT_ASYNC
#define HAVE_WAIT_ASYNC 0
#endif

#if HAVE_ASYNC_B128
#if HAVE_WAIT_ASYNC
#define WAIT_ASYNC(n) __builtin_amdgcn_s_wait_asynccnt(n)
#else
#define WAIT_ASYNC(n) asm volatile("s_wait_asynccnt %0" ::"i"(n) : "memory")
#endif

typedef int v4i __attribute__((ext_vector_type(4)));
typedef __attribute__((address_space(1))) v4i* v4i_gptr;  // global
typedef __attribute__((address_space(3))) v4i* v4i_lptr;  // LDS

__device__ __forceinline__ void async_cp16(const char* g, char* l) {
  __builtin_amdgcn_global_load_async_to_lds_b128(
      (v4i_gptr)(g), (v4i_lptr)(l), /*offset=*/0, /*cpol=*/0);
}
#endif

typedef unsigned long long u64;

// butterfly max-reduce of a packed u64 key across the wave32;
// afterwards ALL lanes hold the maximum key.
__device__ __forceinline__ u64 wave_allmax_u64(u64 key) {
#pragma unroll
  for (int m = 16; m > 0; m >>= 1) {
    unsigned lo = (unsigned)key, hi = (unsigned)(key >> 32);
    lo = __shfl_xor(lo, m, 32);
    hi = __shfl_xor(hi, m, 32);
    u64 ok = ((u64)hi << 32) | lo;
    key = (ok > key) ? ok : key;
  }
  return key;
}

__global__ __launch_bounds__(kT) void fps_kernel(const float* __restrict__ xyz,
                                                 float* __restrict__ out) {
  const int b = blockIdx.x;
  const int tid = threadIdx.x;
  const int lane = tid & 31;
  const int wave = tid >> 5;

  const float* gbase = xyz + (size_t)b * kN * 3;
  float* outb = out + (size_t)b * kM * 3;

  extern __shared__ char smem[];
  char* stage = smem + wave * (kBuf * kChunkBytes);
  u64* redK = (u64*)(smem + kStageBytes);   // per-wave winning key
  float* redX = (float*)(redK + kW);        // per-wave winning coords
  float* redY = redX + kW;
  float* redZ = redY + kW;
  float* ctr = redZ + kW;                   // broadcast center coords [3]

  // running min squared distance for the 64 points owned by this lane;
  // dist[c*4+jj] <-> point index waveBase + c*128 + lane*4 + jj
  float dist[kJ];
#pragma unroll
  for (int j = 0; j < kJ; ++j) dist[j] = 1e10f;

  if (tid == 0) {
    float x0 = gbase[0], y0 = gbase[1], z0 = gbase[2];
    ctr[0] = x0;
    ctr[1] = y0;
    ctr[2] = z0;
    outb[0] = x0;
    outb[1] = y0;
    outb[2] = z0;
  }
  __syncthreads();

  const int waveBase = wave * kPtsPerWave;
  const char* gwave = (const char*)(gbase) + (size_t)waveBase * 12;

  for (int k = 1; k < kM; ++k) {
    const float cx = ctr[0], cy = ctr[1], cz = ctr[2];

    float bv = -1.0f;  // every distance >= 0, so first update always takes
    int bi = waveBase + lane * kChunkJ;

#define FPS_UPDATE(jconst, pidx, px, py, pz)                       \
  do {                                                             \
    float dx_ = (px)-cx, dy_ = (py)-cy, dz_ = (pz)-cz;             \
    float d_ = dx_ * dx_ + dy_ * dy_ + dz_ * dz_;                  \
    float nd_ = fminf(dist[jconst], d_);                           \
    dist[jconst] = nd_;                                            \
    if (nd_ > bv) { /* '>' keeps earliest index within a lane */   \
      bv = nd_;                                                    \
      bi = (pidx);                                                 \
    }                                                              \
  } while (0)

    // consume 4 points (48B = 3 x float4) owned by this lane in chunk c
#define FPS_CONSUME4(cbuf, cconst)                                 \
  do {                                                             \
    const float4* lp_ = (const float4*)(cbuf) + lane * 3;          \
    float4 q0_ = lp_[0], q1_ = lp_[1], q2_ = lp_[2];               \
    const int pb_ = waveBase + (cconst)*kChunkPts + lane * 4;      \
    FPS_UPDATE((cconst)*4 + 0, pb_ + 0, q0_.x, q0_.y, q0_.z);      \
    FPS_UPDATE((cconst)*4 + 1, pb_ + 1, q0_.w, q1_.x, q1_.y);      \
    FPS_UPDATE((cconst)*4 + 2, pb_ + 2, q1_.z, q1_.w, q2_.x);      \
    FPS_UPDATE((cconst)*4 + 3, pb_ + 3, q2_.y, q2_.z, q2_.w);      \
  } while (0)

#if HAVE_ASYNC_B128
    // depth-3 async prefetch: chunks c..c+3 in flight (12 b128 ops max);
    // s_wait_asynccnt <= 9 guarantees chunk c has landed (in-order done).
#define FPS_ISSUE(cc, bufi)                                        \
  do {                                                             \
    const char* gsrc_ = gwave + (cc)*kChunkBytes + lane * 48;      \
    char* ldst_ = stage + (bufi)*kChunkBytes + lane * 48;          \
    async_cp16(gsrc_, ldst_);                                      \
    async_cp16(gsrc_ + 16, ldst_ + 16);                            \
    async_cp16(gsrc_ + 32, ldst_ + 32);                            \
  } while (0)

    FPS_ISSUE(0, 0);
    FPS_ISSUE(1, 1);
    FPS_ISSUE(2, 2);
#pragma unroll
    for (int c = 0; c < kChunks; ++c) {
      if (c + 3 < kChunks) {
        FPS_ISSUE(c + 3, (c + 3) & (kBuf - 1));
        WAIT_ASYNC(9);
      } else if (c == kChunks - 3) {
        WAIT_ASYNC(6);
      } else if (c == kChunks - 2) {
        WAIT_ASYNC(3);
      } else {
        WAIT_ASYNC(0);
      }
      const char* lbuf = stage + (c & (kBuf - 1)) * kChunkBytes;
      FPS_CONSUME4(lbuf, c);
    }
#undef FPS_ISSUE
#else
    // fallback: direct (L2-resident) global streaming, same blocked layout
#pragma unroll
    for (int c = 0; c < kChunks; ++c) {
      const char* gchunk = gwave + c * kChunkBytes;
      FPS_CONSUME4(gchunk, c);
    }
#endif
#undef FPS_CONSUME4
#undef FPS_UPDATE

    // each lane fetches its own candidate's coords (overlapped, L2-hot)
    const float* cp = gbase + (size_t)bi * 3;
    float candx = cp[0], candy = cp[1], candz = cp[2];

    // packed sortable key: bv >= 0 here, so float bits are order-preserving;
    // ~bi makes equal values prefer the lowest index (argmax semantics).
    u64 key = ((u64)__float_as_uint(bv) << 32) | (unsigned)(~bi);
    key = wave_allmax_u64(key);

    // winner's lane: candidate indices satisfy lane == (idx & 127) >> 2
    const int widx = (int)(~(unsigned)key);
    const int wlane = (widx & 127) >> 2;
    float wx = __shfl(candx, wlane, 32);
    float wy = __shfl(candy, wlane, 32);
    float wz = __shfl(candz, wlane, 32);
    if (lane == 0) {
      redK[wave] = key;
      redX[wave] = wx;
      redY[wave] = wy;
      redZ[wave] = wz;
    }
    __syncthreads();

    if (tid < 32) {
      u64 gk = wave_allmax_u64(redK[lane]);
      if (lane == 0) {
        const int gidx = (int)(~(unsigned)gk);
        const int gwaveid = gidx >> 11;  // idx / kPtsPerWave
        float x = redX[gwaveid], y = redY[gwaveid], z = redZ[gwaveid];
        ctr[0] = x;
        ctr[1] = y;
        ctr[2] = z;
        outb[(size_t)k * 3 + 0] = x;
        outb[(size_t)k * 3 + 1] = y;
        outb[(size_t)k * 3 + 2] = z;
      }
    }
    __syncthreads();
  }
}

extern "C" void kernel_launch(void* const* d_in, const int* in_sizes, int n_in,
                              void* d_out, int out_size, void* d_ws,
                              size_t ws_size, hipStream_t stream) {
  (void)n_in;
  (void)out_size;
  (void)d_ws;
  (void)ws_size;
  const float* xyz = (const float*)d_in[0];
  float* out = (float*)d_out;

  const int B = in_sizes[0] / (kN * 3);  // 32
  const size_t shmem = (size_t)kStageBytes + sizeof(u64) * kW +
                       sizeof(float) * kW * 3 + sizeof(float) * 4;

  fps_kernel<<<dim3(B), dim3(kT), shmem, stream>>>(xyz, out);
}